// DandCrecon_89266600280715
// MI455X (gfx1250) — compile-verified
//
#include <hip/hip_runtime.h>
#include <math.h>

// ---- problem constants (match reference) ----
#define N_PROJ 96
#define A2D    96
#define DET2D  192
#define HH     128
#define WW     128
#define DSO_F  500.0f
#define DSD_F  1000.0f
#define KMAG   (1000.0f / 3.5f)   // DSD / DU == DSD / DV

// ---------------------------------------------------------------------------
// CDNA5 async global->LDS copy (ASYNCcnt path). Per-lane: 16 bytes from
// (saddr + vaddr_off) into LDS byte address lds_addr (GVS addressing mode).
// ---------------------------------------------------------------------------
__device__ __forceinline__ void async_copy_b128(unsigned lds_addr,
                                                const void* sbase,
                                                unsigned goff) {
    asm volatile("global_load_async_to_lds_b128 %0, %1, %2"
                 :: "v"(lds_addr), "v"(goff), "s"(sbase)
                 : "memory");
}
__device__ __forceinline__ void wait_asynccnt0() {
    asm volatile("s_wait_asynccnt 0" ::: "memory");
}
__device__ __forceinline__ unsigned lds_addr_of(const void* p) {
    // LDS flat pointers carry the LDS byte offset in the low 32 bits.
    return (unsigned)(size_t)p;
}

// ---------------------------------------------------------------------------
// Kernel A: deriv[p,a,d] = grad_lastdim(sino*weight)
// One 192-thread block per (p,a) row; ws row staged in LDS.
// ---------------------------------------------------------------------------
__global__ __launch_bounds__(DET2D) void k_deriv(const float* __restrict__ sino,
                                                 const float* __restrict__ wgt,
                                                 float* __restrict__ deriv) {
    __shared__ float ws[DET2D];
    const int row = blockIdx.x;           // p*A2D + a
    const int d   = threadIdx.x;
    const size_t base = (size_t)row * DET2D;
    ws[d] = sino[base + d] * wgt[base + d];
    __syncthreads();
    float g;
    if (d == 0) {
        g = ws[1] - ws[0];
    } else if (d == DET2D - 1) {
        g = ws[DET2D - 1] - ws[DET2D - 2];
    } else {
        g = 0.5f * (ws[d + 1] - ws[d - 1]);
    }
    deriv[base + d] = g;
}

// ---------------------------------------------------------------------------
// Kernel B: cbw[p,y,x] = weights3d(y,x) * sum_a lerp(deriv[p,a,:], pos(a,y,x))
// One projection slab (96x192 = 72KB) staged in LDS via async-to-LDS.
// Geometry note: pos = cos*xg + sin*yg + 95.5 is ALWAYS in [5.7, 185.3]
// (|c*xg+s*yg| <= 63.5*sqrt(2) = 89.8), so i0,i0+1 are always in-bounds:
// the reference's masks/clamps are identities and are omitted.
// ---------------------------------------------------------------------------
#define B_SPLIT 4
#define B_PIX   (HH * WW / B_SPLIT)   // 4096 pixels per block
#define B_PPT   (B_PIX / 256)         // 16 pixels per thread

__global__ __launch_bounds__(256) void k_bp2d(const float* __restrict__ deriv,
                                              float* __restrict__ cbw) {
    __shared__ float L[A2D * DET2D];          // 73728 B
    __shared__ float cth[A2D], sth[A2D];

    const int p   = blockIdx.x / B_SPLIT;
    const int seg = blockIdx.x % B_SPLIT;
    const int tid = threadIdx.x;

    if (tid < A2D) {
        float th = (float)tid * (float)(M_PI / (double)A2D);
        float sv, cv;
        sincosf(th, &sv, &cv);
        cth[tid] = cv;
        sth[tid] = sv;
    }

    // Stage deriv[p] into LDS: 18432 floats = 4608 float4 chunks, 18 per thread.
    const unsigned ldsL  = lds_addr_of(&L[0]);
    const unsigned gbase = (unsigned)(p * A2D * DET2D) * 4u;
#pragma unroll
    for (int k = 0; k < 18; ++k) {
        unsigned eb = (unsigned)(tid + k * 256) * 16u;
        async_copy_b128(ldsL + eb, deriv, gbase + eb);
    }
    wait_asynccnt0();
    __syncthreads();

    float xg[B_PPT], yg[B_PPT], acc[B_PPT];
#pragma unroll
    for (int i = 0; i < B_PPT; ++i) {
        int pix = seg * B_PIX + tid + i * 256;
        int y = pix >> 7, x = pix & 127;
        xg[i]  = (float)x - 63.5f;
        yg[i]  = (float)y - 63.5f;
        acc[i] = 0.0f;
    }

    for (int a = 0; a < A2D; ++a) {
        const float c = cth[a], s = sth[a];
        const float* Lr = L + a * DET2D;
#pragma unroll
        for (int i = 0; i < B_PPT; ++i) {
            float pos = fmaf(c, xg[i], fmaf(s, yg[i], 95.5f));
            float fp  = floorf(pos);
            int   i0  = (int)fp;
            float f   = pos - fp;
            float v0  = Lr[i0];
            float v1  = Lr[i0 + 1];
            acc[i] += fmaf(f, v1 - v0, v0);
        }
    }

    float* out = cbw + (size_t)p * (HH * WW);
#pragma unroll
    for (int i = 0; i < B_PPT; ++i) {
        int pix = seg * B_PIX + tid + i * 256;
        float w3 = DSD_F * rsqrtf(DSD_F * DSD_F + xg[i] * xg[i] + yg[i] * yg[i]);
        out[pix] = acc[i] * w3;
    }
}

// ---------------------------------------------------------------------------
// Kernel C: cone-beam bp3d over 96 betas + PReLU.
// Each beta's 128x128 projection (64KB) staged in a double-buffered LDS
// region (128KB — enabled by CDNA5's 320KB LDS) via async-to-LDS; beta+1 is
// prefetched while beta is consumed. 1024 blocks x 256 threads, 8 voxels per
// thread accumulated in registers.
// Geometry note: pu in [11.3,115.7], pv in [19.3,107.7] for all voxels/betas
// (|s|/r <= 89.8/sqrt(500^2-89.8^2)), so the reference's masks/clamps are
// identities and are omitted; bilinear is computed in lerp form.
// ---------------------------------------------------------------------------
__global__ __launch_bounds__(256) void k_bp3d(const float* __restrict__ cbw,
                                              const float* __restrict__ prelu,
                                              float* __restrict__ out) {
    __shared__ float buf[2][HH * WW];         // 2 x 65536 B
    __shared__ float cbt[N_PROJ], sbt[N_PROJ];

    const int tid = threadIdx.x;
    if (tid < N_PROJ) {
        float b = (float)tid * (float)(2.0 * M_PI / (double)N_PROJ);
        float sv, cv;
        sincosf(b, &sv, &cv);
        cbt[tid] = cv;
        sbt[tid] = sv;
    }

    const int z    = blockIdx.x >> 3;          // 8 blocks per z-slice
    const int y0   = (blockIdx.x & 7) * 16;    // 16 y-rows per block
    const int x    = tid & 127;
    const int ysub = tid >> 7;

    const float xx = (float)x - 63.5f;
    const float zz = (float)z - 63.5f;
    const float qv = KMAG * zz;                // hoisted pv numerator

    float yy[8], acc[8];
#pragma unroll
    for (int k = 0; k < 8; ++k) {
        yy[k]  = (float)(y0 + ysub + 2 * k) - 63.5f;
        acc[k] = 0.0f;
    }

    const unsigned lds0 = lds_addr_of(&buf[0][0]);
    const unsigned lds1 = lds_addr_of(&buf[1][0]);

    // Prologue: stage beta=0 into buffer 0 (16384 floats = 16 b128/thread).
#pragma unroll
    for (int k = 0; k < 16; ++k) {
        unsigned eb = (unsigned)(tid + k * 256) * 16u;
        async_copy_b128(lds0 + eb, cbw, eb);
    }
    wait_asynccnt0();
    __syncthreads();

    for (int beta = 0; beta < N_PROJ; ++beta) {
        // Prefetch next projection into the other buffer (async, overlapped).
        if (beta + 1 < N_PROJ) {
            unsigned ldsn = (beta & 1) ? lds0 : lds1;
            unsigned gb   = (unsigned)((beta + 1) * HH * WW) * 4u;
#pragma unroll
            for (int k = 0; k < 16; ++k) {
                unsigned eb = (unsigned)(tid + k * 256) * 16u;
                async_copy_b128(ldsn + eb, cbw, gb + eb);
            }
        }

        const float* P  = buf[beta & 1];
        const float  cb = cbt[beta];
        const float  sb = sbt[beta];

#pragma unroll
        for (int k = 0; k < 8; ++k) {
            float t  = fmaf(xx, cb, yy[k] * sb);
            float s2 = fmaf(yy[k], cb, -xx * sb);
            float r  = DSO_F + t;
            float ir = __builtin_amdgcn_rcpf(r);      // v_rcp_f32 (1 ulp)
            float pu = fmaf(KMAG * s2, ir, 63.5f);
            float pv = fmaf(qv, ir, 63.5f);

            float fu0 = floorf(pu); int iu0 = (int)fu0; float fu = pu - fu0;
            float fv0 = floorf(pv); int iv0 = (int)fv0; float fv = pv - fv0;

            int idx = iv0 * WW + iu0;
            float p00 = P[idx];
            float p01 = P[idx + 1];
            float p10 = P[idx + WW];
            float p11 = P[idx + WW + 1];

            float a0  = fmaf(fu, p01 - p00, p00);
            float a1  = fmaf(fu, p11 - p10, p10);
            float val = fmaf(fv, a1 - a0, a0);

            float w2 = (DSD_F * DSD_F) * (ir * ir);   // (DSD/r)^2
            acc[k] = fmaf(val, w2, acc[k]);
        }

        wait_asynccnt0();   // next buffer fully resident (this wave)
        __syncthreads();    // ... for all waves; also fences buffer reuse
    }

    const float a = prelu[0];
#pragma unroll
    for (int k = 0; k < 8; ++k) {
        float v = acc[k];
        int y = y0 + ysub + 2 * k;
        out[(size_t)z * (HH * WW) + y * WW + x] = (v >= 0.0f) ? v : a * v;
    }
}

// ---------------------------------------------------------------------------
// Host-side launch
// ---------------------------------------------------------------------------
extern "C" void kernel_launch(void* const* d_in, const int* in_sizes, int n_in,
                              void* d_out, int out_size, void* d_ws, size_t ws_size,
                              hipStream_t stream) {
    const float* sino  = (const float*)d_in[0];  // (1,1,96,96,192) f32
    const float* wgt   = (const float*)d_in[1];  // (1,96,96,192)   f32
    const float* prelu = (const float*)d_in[2];  // (1,)            f32
    float*       out   = (float*)d_out;          // (1,128,128,128) f32

    float* deriv = (float*)d_ws;                          // 96*96*192 floats
    float* cbw   = deriv + (size_t)N_PROJ * A2D * DET2D;  // 96*128*128 floats

    k_deriv<<<N_PROJ * A2D, DET2D, 0, stream>>>(sino, wgt, deriv);
    k_bp2d <<<N_PROJ * B_SPLIT, 256, 0, stream>>>(deriv, cbw);
    k_bp3d <<<(HH * WW * 128) / (256 * 8), 256, 0, stream>>>(cbw, prelu, out);
}